// Mnn_Layer_without_Rho_60301340836048
// MI455X (gfx1250) — compile-verified
//
#include <hip/hip_runtime.h>
#include <math.h>

typedef __attribute__((ext_vector_type(2))) float v2f;
typedef __attribute__((ext_vector_type(4))) float v4f;
typedef __attribute__((ext_vector_type(8))) float v8f;

#define B_ROWS 512
#define NDIM   2048
#define KDIM   2048

// ---- workspace layout (in floats) ----
#define WS_OUT1   0
#define WS_OUT2SQ (B_ROWS * NDIM)              // out2^2 (variance-path GEMM result)
#define WS_MU     (2 * B_ROWS * NDIM)
#define WS_RVB    (WS_MU + NDIM)               // rsqrt(var_biased + eps)
#define WS_SU     (WS_RVB + NDIM)              // bn_w^2 / (var_unbiased + eps)
#define WS_GLX    (WS_SU + NDIM)
#define WS_GLW    (WS_GLX + 64)
#define WS_CUMQ   (WS_GLW + 64)
#define NQ 2001

#define SQRT_PI_HALF 0.886226925452758f        // 0.5*sqrt(pi)
#define SQRT_L       0.223606797749979f        // sqrt(0.05)

// ---------------------------------------------------------------------------
// Async global->LDS copy (CDNA5 Tensor/Async path, tracked with ASYNCcnt).
// vdst = per-lane LDS byte address, vaddr = per-lane 64-bit global address.
// ---------------------------------------------------------------------------
__device__ __forceinline__ void async_ld16(unsigned lds_addr, const float* gptr) {
    asm volatile("global_load_async_to_lds_b128 %0, %1, off"
                 :: "v"(lds_addr), "v"(gptr) : "memory");
}
__device__ __forceinline__ void wait_async0() {
    asm volatile("s_wait_asynccnt 0" ::: "memory");
}
__device__ __forceinline__ unsigned lds_off(const void* p) {
    // generic LDS pointer: low 32 bits are the LDS byte address
    return (unsigned)(unsigned long long)p;
}

// ---------------------------------------------------------------------------
// Kernel 0: build Gauss-Legendre(40) nodes/weights (fp64 Newton) and the
// cumulative inner-Dawson table CUM_Q on the uniform grid linspace(-8,4,2001).
// ---------------------------------------------------------------------------
__global__ __launch_bounds__(256) void init_tables(float* __restrict__ ws) {
    const int t = threadIdx.x;
    if (t < 40) {
        const int n = 40;
        const double PI = 3.14159265358979323846;
        double x = cos(PI * ((double)t + 0.75) / ((double)n + 0.5));
        double pp = 1.0;
        for (int it = 0; it < 48; ++it) {
            double p0 = 1.0, p1 = x;
            for (int j = 2; j <= n; ++j) {
                double p2 = ((2.0 * j - 1.0) * x * p1 - (j - 1.0) * p0) / (double)j;
                p0 = p1; p1 = p2;
            }
            pp = (double)n * (x * p1 - p0) / (x * x - 1.0);
            x -= p1 / pp;
        }
        ws[WS_GLX + t] = (float)x;
        ws[WS_GLW + t] = (float)(2.0 / ((1.0 - x * x) * pp * pp));
    }
    // q(y) = exp(-y^2) * g1(y)^2 on the fixed grid
    const float dy = 12.0f / 2000.0f;
    for (int i = t; i < NQ; i += 256) {
        float y = -8.0f + dy * (float)i;
        float g = SQRT_PI_HALF * expf(y * y) * erfcf(-y);
        ws[WS_CUMQ + i] = expf(-y * y) * g * g;
    }
    __syncthreads();
    if (t == 0) {
        float prev = ws[WS_CUMQ + 0];
        float c = 0.0f;
        ws[WS_CUMQ + 0] = 0.0f;
        for (int i = 1; i < NQ; ++i) {
            float cur = ws[WS_CUMQ + i];
            c += 0.5f * (cur + prev) * dy;
            ws[WS_CUMQ + i] = c;
            prev = cur;
        }
    }
}

// ---------------------------------------------------------------------------
// Kernel 1: fused dual GEMM with V_WMMA_F32_16X16X4_F32.
//   out1   = ubar     @ W^T
//   out2sq = (sbar^2) @ (W^2)^T          (W loaded once; squared in registers)
// Block = 256 threads (8 waves), tile 64(M) x 128(N).
// Double-buffered LDS stages filled with GLOBAL_LOAD_ASYNC_TO_LDS_B128
// (ASYNCcnt) so stage s+1 global traffic overlaps stage s WMMAs.
// Wave (wm 0..3, wn 0..1) owns a 16x64 strip -> 4 N-tiles x 2 GEMMs.
// ---------------------------------------------------------------------------
#define TM 64
#define TN 128
#define TK 16
#define LDT 20   // padded row stride: conflict-free, keeps 16B alignment

__global__ __launch_bounds__(256)
void dual_gemm(const float* __restrict__ ubar, const float* __restrict__ sbar,
               const float* __restrict__ weight, float* __restrict__ ws) {
    __shared__ float lA[2][TM][LDT];
    __shared__ float lS[2][TM][LDT];
    __shared__ float lW[2][TN][LDT];

    float* __restrict__ out1 = ws + WS_OUT1;
    float* __restrict__ out2 = ws + WS_OUT2SQ;

    const int t    = threadIdx.x;
    const int lane = t & 31;
    const int wid  = t >> 5;
    const int wm   = wid & 3;          // M strip (16 rows)
    const int wn   = wid >> 2;         // N strip (64 cols)
    const int l16  = lane & 15;
    const int lh   = lane >> 4;        // half-wave select
    const int m0   = blockIdx.y * TM;
    const int n0   = blockIdx.x * TN;
    const int lr   = t >> 2;           // tile row staged by this thread (0..63)
    const int lc   = (t & 3) * 4;      // 4 consecutive K elements

    const float* ga  = &ubar[(m0 + lr) * KDIM + lc];
    const float* gs  = &sbar[(m0 + lr) * KDIM + lc];
    const float* gw0 = &weight[(n0 + lr) * KDIM + lc];
    const float* gw1 = &weight[(n0 + lr + 64) * KDIM + lc];

    v8f acc1[4] = {{}, {}, {}, {}};
    v8f acc2[4] = {{}, {}, {}, {}};

    const int NS = KDIM / TK;   // 128 stages

    // prologue: stage 0 into buffer 0
    async_ld16(lds_off(&lA[0][lr][lc]), ga);
    async_ld16(lds_off(&lS[0][lr][lc]), gs);
    async_ld16(lds_off(&lW[0][lr][lc]), gw0);
    async_ld16(lds_off(&lW[0][lr + 64][lc]), gw1);

    for (int s = 0; s < NS; ++s) {
        wait_async0();          // this wave's stage-s fills are in LDS
        __syncthreads();        // publish to all waves / retire reads of other buffer
        if (s + 1 < NS) {
            const int nb = (s + 1) & 1;
            const int kb = (s + 1) * TK;
            async_ld16(lds_off(&lA[nb][lr][lc]), ga + kb);
            async_ld16(lds_off(&lS[nb][lr][lc]), gs + kb);
            async_ld16(lds_off(&lW[nb][lr][lc]), gw0 + kb);
            async_ld16(lds_off(&lW[nb][lr + 64][lc]), gw1 + kb);
        }
        const int buf = s & 1;
        const int ar  = wm * 16 + l16;
#pragma unroll
        for (int kk = 0; kk < 4; ++kk) {
            const int ko = kk * 4 + lh * 2;   // K=0,1 (lanes 0-15) / K=2,3 (lanes 16-31)
            v2f a  = *(const v2f*)&lA[buf][ar][ko];
            v2f sv = *(const v2f*)&lS[buf][ar][ko];
            v2f s2 = sv * sv;
#pragma unroll
            for (int j = 0; j < 4; ++j) {
                v2f b  = *(const v2f*)&lW[buf][wn * 64 + j * 16 + l16][ko];
                v2f b2 = b * b;
                acc1[j] = __builtin_amdgcn_wmma_f32_16x16x4_f32(
                    false, a,  false, b,  (short)0, acc1[j], false, false);
                acc2[j] = __builtin_amdgcn_wmma_f32_16x16x4_f32(
                    false, s2, false, b2, (short)0, acc2[j], false, false);
            }
        }
    }

    // D layout: lane<16 -> rows m0+v, lane>=16 -> rows m0+8+v (v = vgpr index)
    const int rbase = m0 + wm * 16 + lh * 8;
#pragma unroll
    for (int j = 0; j < 4; ++j) {
        const int col = n0 + wn * 64 + j * 16 + l16;
#pragma unroll
        for (int v = 0; v < 8; ++v) {
            out1[(rbase + v) * NDIM + col] = acc1[j][v];
            out2[(rbase + v) * NDIM + col] = acc2[j][v];
        }
    }
}

// ---------------------------------------------------------------------------
// Kernel 2: per-column batch statistics of out1.
// ---------------------------------------------------------------------------
__global__ __launch_bounds__(256)
void col_stats(const float* __restrict__ bn_weight, float* __restrict__ ws) {
    __shared__ float ssum[256];
    __shared__ float ssq[256];
    const int n = blockIdx.x;
    const float* out1 = ws + WS_OUT1;
    float a = out1[threadIdx.x * NDIM + n];
    float b = out1[(threadIdx.x + 256) * NDIM + n];
    ssum[threadIdx.x] = a + b;
    ssq[threadIdx.x]  = a * a + b * b;
    __syncthreads();
    for (int off = 128; off > 0; off >>= 1) {
        if (threadIdx.x < off) {
            ssum[threadIdx.x] += ssum[threadIdx.x + off];
            ssq[threadIdx.x]  += ssq[threadIdx.x + off];
        }
        __syncthreads();
    }
    if (threadIdx.x == 0) {
        float sum = ssum[0], sq = ssq[0];
        float mu   = sum * (1.0f / 512.0f);
        float varb = sq * (1.0f / 512.0f) - mu * mu;             // biased (BN)
        float varu = (sq - 512.0f * mu * mu) * (1.0f / 511.0f);  // unbiased
        float bw = bn_weight[n];
        ws[WS_MU  + n] = mu;
        ws[WS_RVB + n] = rsqrtf(varb + 1e-5f);
        ws[WS_SU  + n] = bw * bw / (varu + 1e-5f);
    }
}

// ---------------------------------------------------------------------------
// Kernel 3: LIF moment activation (40-node Gauss-Legendre).
// ---------------------------------------------------------------------------
__global__ __launch_bounds__(256)
void moment_act(const float* __restrict__ bn_weight, const float* __restrict__ bn_bias,
                const float* __restrict__ ext_bias, const float* __restrict__ ws,
                float* __restrict__ out) {
    __shared__ float glx[40];
    __shared__ float glw[40];
    __shared__ float cq[NQ];
    for (int i = threadIdx.x; i < NQ; i += 256) cq[i] = ws[WS_CUMQ + i];
    if (threadIdx.x < 40) {
        glx[threadIdx.x] = ws[WS_GLX + threadIdx.x];
        glw[threadIdx.x] = ws[WS_GLW + threadIdx.x];
    }
    __syncthreads();

    const int idx = blockIdx.x * 256 + threadIdx.x;
    const int n   = idx & (NDIM - 1);

    float o1   = ws[WS_OUT1   + idx];
    float o2sq = ws[WS_OUT2SQ + idx];
    float uhat = (o1 - ws[WS_MU + n]) * ws[WS_RVB + n] * bn_weight[n] + bn_bias[n];
    float eb   = ext_bias[n];
    float shat = sqrtf(o2sq * ws[WS_SU + n] + eb * eb);
    float inv  = 1.0f / (shat * SQRT_L);
    float ub   = fminf(fmaxf((1.0f - uhat) * inv, -4.0f), 4.0f);  // VTH*L = 1
    float lb   = fminf(fmaxf((-uhat) * inv, -4.0f), 4.0f);        // VRES*L = 0
    float c  = 0.5f * (ub + lb);
    float hw = 0.5f * (ub - lb);

    float aT = 0.0f, aV = 0.0f;
#pragma unroll 4
    for (int i = 0; i < 40; ++i) {
        float x   = c + hw * glx[i];
        float w   = glw[i];
        float ex2 = expf(x * x);
        float g   = SQRT_PI_HALF * ex2 * erfcf(-x);               // g1(x)
        float tp  = fminf(fmaxf((x + 8.0f) * (2000.0f / 12.0f), 0.0f), 2000.0f);
        int   i0  = (int)tp; if (i0 > 1999) i0 = 1999;
        float fr  = tp - (float)i0;
        float hv  = ex2 * (cq[i0] + fr * (cq[i0 + 1] - cq[i0]));  // h2(x)
        aT += g * w;
        aV += hv * w;
    }
    float T    = 40.0f  * aT * hw;     // (2/L) * integral
    float u    = 1.0f / (5.0f + T);    // 1/(T_REF + T)
    float varT = 3200.0f * aV * hw;    // (8/L^2) * integral
    float s    = sqrtf(fmaxf(varT, 0.0f)) * u * sqrtf(u);

    out[idx] = u;
    out[B_ROWS * NDIM + idx] = s;
}

// ---------------------------------------------------------------------------
extern "C" void kernel_launch(void* const* d_in, const int* in_sizes, int n_in,
                              void* d_out, int out_size, void* d_ws, size_t ws_size,
                              hipStream_t stream) {
    const float* ubar      = (const float*)d_in[0];
    const float* sbar      = (const float*)d_in[1];
    const float* weight    = (const float*)d_in[2];
    const float* bn_weight = (const float*)d_in[3];
    const float* bn_bias   = (const float*)d_in[4];
    const float* ext_bias  = (const float*)d_in[5];
    float* ws  = (float*)d_ws;
    float* out = (float*)d_out;

    init_tables<<<1, 256, 0, stream>>>(ws);
    dim3 grid(NDIM / TN, B_ROWS / TM);
    dual_gemm<<<grid, 256, 0, stream>>>(ubar, sbar, weight, ws);
    col_stats<<<NDIM, 256, 0, stream>>>(bn_weight, ws);
    moment_act<<<(B_ROWS * NDIM) / 256, 256, 0, stream>>>(bn_weight, bn_bias, ext_bias, ws, out);
}